// NumToLang_35055523070755
// MI455X (gfx1250) — compile-verified
//
#include <hip/hip_runtime.h>

#define MAXLEN 5

typedef float v2f __attribute__((ext_vector_type(2)));
typedef float v4f __attribute__((ext_vector_type(4)));
typedef float v8f __attribute__((ext_vector_type(8)));

// D = A(16x4 f32) x B(4x16 f32) + C, full fp32 WMMA (VOP3P op 93)
__device__ __forceinline__ v8f wmma4(v2f a, v2f b, v8f c) {
  return __builtin_amdgcn_wmma_f32_16x16x4_f32(false, a, false, b, (short)0, c,
                                               false, false);
}

__device__ __forceinline__ float sigm(float x) {
  return __builtin_amdgcn_rcpf(1.0f + __expf(-x));
}
__device__ __forceinline__ float tanhfast(float x) {
  float r = __builtin_amdgcn_rcpf(1.0f + __expf(-2.0f * x));
  return 2.0f * r - 1.0f;
}
__device__ __forceinline__ float swap16(float x) { return __shfl_xor(x, 16, 32); }

// Load one K=4 chunk of a weight matrix (G x K, row-major) into WMMA A-format:
// lane L holds row (L%16), K elements {4c+2*half, 4c+2*half+1}. Pad with 0.
__device__ __forceinline__ v2f loadA(const float* __restrict__ W, int G, int K,
                                     int chunk, int gate, int half) {
  int k0 = chunk * 4 + half * 2;
  v2f r;
  r.x = (gate < G && (k0 + 0) < K) ? W[gate * K + k0 + 0] : 0.0f;
  r.y = (gate < G && (k0 + 1) < K) ? W[gate * K + k0 + 1] : 0.0f;
  return r;
}

__global__ __launch_bounds__(256) void numtolang_rnn(
    const float* __restrict__ x, const float* __restrict__ onehots,
    const float* __restrict__ W1, const float* __restrict__ b1,
    const float* __restrict__ gwi, const float* __restrict__ gwh,
    const float* __restrict__ gbi, const float* __restrict__ gbh,
    const float* __restrict__ lwi, const float* __restrict__ lwh,
    const float* __restrict__ lbi, const float* __restrict__ lbh,
    const float* __restrict__ W2, const float* __restrict__ b2,
    float* __restrict__ out, int B)
{
  const int lane = threadIdx.x & 31;
  const int wv   = threadIdx.x >> 5;
  const int half = lane >> 4;      // 0: gates 0-7 / K 0-1, 1: gates 8-15 / K 2-3
  const int ln   = lane & 15;      // batch column within the wave's 16-row tile
  const long tile = ((long)blockIdx.x * (blockDim.x >> 5) + wv) * 16;
  if (tile >= (long)B) return;     // wave-uniform exit; EXEC all-ones below
  const long row   = tile + ln;
  const bool rowok = row < (long)B;
  const long rc    = rowok ? row : (long)B - 1;

  // ---- Preload all weights in WMMA A-format (uniform across waves) ----
  v2f Agwi0 = loadA(gwi, 12, 12, 0, ln, half);
  v2f Agwi1 = loadA(gwi, 12, 12, 1, ln, half);
  v2f Agwi2 = loadA(gwi, 12, 12, 2, ln, half);
  v2f Agwh  = loadA(gwh, 12,  4, 0, ln, half);
  v2f Alwi0 = loadA(lwi, 16, 12, 0, ln, half);
  v2f Alwi1 = loadA(lwi, 16, 12, 1, ln, half);
  v2f Alwi2 = loadA(lwi, 16, 12, 2, ln, half);
  v2f Alwh  = loadA(lwh, 16,  4, 0, ln, half);
  v2f AW2   = loadA(W2,  12,  4, 0, ln, half);

  // ---- Biases pre-broadcast into C/D-format (VGPR j = gate j + 8*half) ----
  v8f biD, bhD, blD, b2D;
#pragma unroll
  for (int j = 0; j < 8; ++j) {
    int g = j + 8 * half;
    biD[j] = (g < 12) ? gbi[g] : 0.0f;
    bhD[j] = (g < 12) ? gbh[g] : 0.0f;
    blD[j] = lbi[g] + lbh[g];            // LSTM biases always summed
    b2D[j] = (g < 12) ? b2[g] : 0.0f;
  }

  // ---- h0 = materials @ W1.T + b1 (scalar; once per element) ----
  float xv = x[rc];
  float m0 = xv, m1 = 10.0f * xv, m2 = xv * xv, m3 = m2 * xv, m4 = 2.0f * xv;
  int r0i = 2 * half;                    // this lane owns h[r0i], h[r0i+1]
  const float* wa = W1 + r0i * 5;
  const float* wb = W1 + (r0i + 1) * 5;
  float s0 = b1[r0i]     + wa[0]*m0 + wa[1]*m1 + wa[2]*m2 + wa[3]*m3 + wa[4]*m4;
  float s1 = b1[r0i + 1] + wb[0]*m0 + wb[1]*m1 + wb[2]*m2 + wb[3]*m3 + wb[4]*m4;
  float c0 = 0.0f, c1 = 0.0f;            // LSTM cell state, same split as h

  const float* ohrow = onehots + rc * (MAXLEN * 12) + half * 2;

#pragma unroll
  for (int t = 0; t < MAXLEN; ++t) {
    // ---- inputs in B-format: lane n holds inp[n][4c+2h .. 4c+2h+1] ----
    v2f B0 = {0.0f, 0.0f}, B1 = {0.0f, 0.0f}, B2 = {0.0f, 0.0f};
    if (t > 0) {
      const float* p = ohrow + (t - 1) * 12;
      B0 = *(const v2f*)(p);
      B1 = *(const v2f*)(p + 4);
      B2 = *(const v2f*)(p + 8);
    }
    v2f Bh; Bh.x = s0; Bh.y = s1;        // h already in B-format

    // ---- GRU: gi = wi*inp + bi ; gh = wh*h + bh (gate-major) ----
    v8f gi = biD;
    gi = wmma4(Agwi0, B0, gi);
    gi = wmma4(Agwi1, B1, gi);
    gi = wmma4(Agwi2, B2, gi);
    v8f gh = bhD;
    gh = wmma4(Agwh, Bh, gh);

    float r0 = sigm(gi[0] + gh[0]);      // valid in lower half (gates 0-3)
    float r1 = sigm(gi[1] + gh[1]);
    float r2 = sigm(gi[2] + gh[2]);
    float r3 = sigm(gi[3] + gh[3]);
    float r0u = swap16(r0), r1u = swap16(r1), r2u = swap16(r2), r3u = swap16(r3);
    float n0 = tanhfast(gi[0] + r0u * gh[0]);  // valid in upper half (gates 8-11)
    float n1 = tanhfast(gi[1] + r1u * gh[1]);
    float n2 = tanhfast(gi[2] + r2u * gh[2]);
    float n3 = tanhfast(gi[3] + r3u * gh[3]);
    float z0 = sigm(gi[4] + gh[4]);      // valid in lower half (gates 4-7)
    float z1 = sigm(gi[5] + gh[5]);
    float z2 = sigm(gi[6] + gh[6]);
    float z3 = sigm(gi[7] + gh[7]);
    float n0x = swap16(n0), n1x = swap16(n1);
    float z2x = swap16(z2), z3x = swap16(z3);
    float zz0 = half ? z2x : z0;
    float zz1 = half ? z3x : z1;
    float nn0 = half ? n2 : n0x;
    float nn1 = half ? n3 : n1x;
    s0 = (1.0f - zz0) * nn0 + zz0 * s0;
    s1 = (1.0f - zz1) * nn1 + zz1 * s1;

    // ---- LSTM: g = wi*inp + wh*h_new + (bi+bh) ----
    v2f Bh2; Bh2.x = s0; Bh2.y = s1;
    v8f g = blD;
    g = wmma4(Alwi0, B0, g);
    g = wmma4(Alwi1, B1, g);
    g = wmma4(Alwi2, B2, g);
    g = wmma4(Alwh, Bh2, g);

    float sg0 = sigm(g[0]), sg1 = sigm(g[1]), sg2 = sigm(g[2]), sg3 = sigm(g[3]);
    float sg4 = sigm(g[4]), sg5 = sigm(g[5]), sg6 = sigm(g[6]), sg7 = sigm(g[7]);
    float th0 = tanhfast(g[0]), th1 = tanhfast(g[1]);
    float th2 = tanhfast(g[2]), th3 = tanhfast(g[3]);
    float th0x = swap16(th0), th1x = swap16(th1);
    float sg2x = swap16(sg2), sg3x = swap16(sg3);
    float sg4x = swap16(sg4), sg5x = swap16(sg5);
    float sg6x = swap16(sg6), sg7x = swap16(sg7);
    float F0 = half ? sg6x : sg4;        // forget gate (gates 4-7)
    float F1 = half ? sg7x : sg5;
    float I0 = half ? sg2x : sg0;        // input gate (gates 0-3)
    float I1 = half ? sg3x : sg1;
    float G0 = half ? th2 : th0x;        // g gate (gates 8-11, tanh)
    float G1 = half ? th3 : th1x;
    float O0 = half ? sg6 : sg4x;        // output gate (gates 12-15)
    float O1 = half ? sg7 : sg5x;
    c0 = F0 * c0 + I0 * G0;
    c1 = F1 * c1 + I1 * G1;
    s0 = O0 * tanhfast(c0);
    s1 = O1 * tanhfast(c1);

    // ---- output head: logits = W2*h + b2, softmax over 12 ----
    v2f Bh3; Bh3.x = s0; Bh3.y = s1;
    v8f lg = b2D;
    lg = wmma4(AW2, Bh3, lg);
    float m03 = fmaxf(fmaxf(lg[0], lg[1]), fmaxf(lg[2], lg[3]));
    float m47 = fmaxf(fmaxf(lg[4], lg[5]), fmaxf(lg[6], lg[7]));
    float mloc = half ? m03 : fmaxf(m03, m47);   // upper half only logits 8-11
    float mall = fmaxf(mloc, swap16(mloc));
    float e0 = __expf(lg[0] - mall), e1 = __expf(lg[1] - mall);
    float e2 = __expf(lg[2] - mall), e3 = __expf(lg[3] - mall);
    float e4 = __expf(lg[4] - mall), e5 = __expf(lg[5] - mall);
    float e6 = __expf(lg[6] - mall), e7 = __expf(lg[7] - mall);
    float slo = (e0 + e1) + (e2 + e3);
    float shi = (e4 + e5) + (e6 + e7);
    float ssum = half ? slo : (slo + shi);
    float stot = ssum + swap16(ssum);
    float inv = __builtin_amdgcn_rcpf(stot);
    if (rowok) {
      float* po = out + ((long)t * B + row) * 12 + half * 8;
      v4f o0 = {e0 * inv, e1 * inv, e2 * inv, e3 * inv};
      *(v4f*)po = o0;                    // lower: v 0-3, upper: v 8-11
      if (!half) {
        v4f o1 = {e4 * inv, e5 * inv, e6 * inv, e7 * inv};
        *(v4f*)(po + 4) = o1;            // lower: v 4-7
      }
    }
  }
}

extern "C" void kernel_launch(void* const* d_in, const int* in_sizes, int n_in,
                              void* d_out, int out_size, void* d_ws, size_t ws_size,
                              hipStream_t stream) {
  const float* x   = (const float*)d_in[0];
  const float* oh  = (const float*)d_in[1];
  const float* W1  = (const float*)d_in[2];
  const float* b1  = (const float*)d_in[3];
  const float* gwi = (const float*)d_in[4];
  const float* gwh = (const float*)d_in[5];
  const float* gbi = (const float*)d_in[6];
  const float* gbh = (const float*)d_in[7];
  const float* lwi = (const float*)d_in[8];
  const float* lwh = (const float*)d_in[9];
  const float* lbi = (const float*)d_in[10];
  const float* lbh = (const float*)d_in[11];
  const float* W2  = (const float*)d_in[12];
  const float* b2  = (const float*)d_in[13];
  int B = in_sizes[0];                 // x is (B, 1)

  const int wavesPerBlock = 256 / 32;  // 8 waves, 16 batch rows each
  long tiles = ((long)B + 15) / 16;
  int blocks = (int)((tiles + wavesPerBlock - 1) / wavesPerBlock);
  numtolang_rnn<<<blocks, 256, 0, stream>>>(x, oh, W1, b1, gwi, gwh, gbi, gbh,
                                            lwi, lwh, lbi, lbh, W2, b2,
                                            (float*)d_out, B);
}